// RankingModel_v2_25366076850461
// MI455X (gfx1250) — compile-verified
//
#include <hip/hip_runtime.h>
#include <hip/hip_bf16.h>
#include <math.h>

typedef __attribute__((ext_vector_type(2)))  float v2f;
typedef __attribute__((ext_vector_type(8)))  float v8f;
typedef unsigned int u32x4 __attribute__((ext_vector_type(4)));
typedef int          i32x4 __attribute__((ext_vector_type(4)));
typedef int          i32x8 __attribute__((ext_vector_type(8)));

#define KDIM 128            // col_num * dmodel
#define HID  32
#define ROWS_PER_BLOCK 64   // 64x128 f32 tile per block
#define TPB_A 128           // 4 waves of 32, one 16-row WMMA tile each
#define LDS_PAD 4           // dwords of pad per row (must match TDM pad_amount)
#define LDS_STRIDE (KDIM + LDS_PAD)
#define NROWS 2048
#define TPB_B 1024

// ---------------------------------------------------------------------------
// Kernel 1: scores = relu(x @ W1 + b1) @ W2 + b2
//   - tile staged global->LDS by the Tensor Data Mover (tensor_load_to_lds)
//     with hardware row padding (pad every 128 dwords by 4 dwords)
//   - exact f32 matmul via v_wmma_f32_16x16x4_f32
// ---------------------------------------------------------------------------
__global__ __launch_bounds__(TPB_A)
void mlp_scores_wmma(const float* __restrict__ table,
                     const float* __restrict__ W1,
                     const float* __restrict__ b1,
                     const float* __restrict__ W2,
                     const float* __restrict__ b2,
                     float* __restrict__ scores,
                     int total_rows)
{
    __shared__ __align__(16) float sA[ROWS_PER_BLOCK][LDS_STRIDE]; // TDM-padded tile
    __shared__ float sB[KDIM][HID];                                // W1 (f32)
    __shared__ float sb1[HID];
    __shared__ float sW2[HID];

    const int tid = threadIdx.x;
    const int r0  = blockIdx.x * ROWS_PER_BLOCK;

    // ---- TDM descriptor (D#) for the 64x128 f32 tile, built per ISA bitfields.
    // Issued from wave 0 only (TENSOR_LOAD_TO_LDS ignores EXEC; one DMA per block).
    if (tid < 32) {
        const unsigned long long gaddr = (unsigned long long)(uintptr_t)(table + (size_t)r0 * KDIM);
        const unsigned int       laddr = (unsigned int)(uintptr_t)&sA[0][0]; // low 32b of flat = LDS offset

        u32x4 g0;
        g0[0] = 1u;                                        // count=1 (valid), user mode
        g0[1] = laddr;                                     // lds_addr
        g0[2] = (unsigned int)gaddr;                       // global_addr[31:0]
        g0[3] = (unsigned int)((gaddr >> 32) & 0x1FFFFFFu) // global_addr[56:32]
              | (2u << 30);                                // type=2 ("image")

        i32x8 g1;
        g1[0] = (2 << 16)        // data_size = 4 bytes
              | (1 << 20)        // pad_enable
              | (6 << 22)        // pad_interval code 6 = every 128 dwords
              | (3 << 25);       // pad_amount  code 3 = 4 dwords
        g1[1] = (KDIM & 0xFFFF) << 16;         // tensor_dim0[15:0] = 128
        g1[2] = (total_rows & 0xFFFF) << 16;   // tensor_dim0[31:16]=0 | tensor_dim1[15:0]
        g1[3] = (KDIM & 0xFFFF) << 16;         // tensor_dim1[31:16]=0 | tile_dim0 = 128
        g1[4] = ROWS_PER_BLOCK & 0xFFFF;       // tile_dim1 = 64 | tile_dim2 = 0
        g1[5] = KDIM;                          // tensor_dim0_stride[31:0] = 128 elements
        g1[6] = 0;                             // stride[47:32]=0 | tensor_dim1_stride lo
        g1[7] = 0;

        i32x4 g2 = {0, 0, 0, 0};               // 2D tensor: groups 2/3 unused
        i32x4 g3 = {0, 0, 0, 0};
#if __clang_major__ >= 23
        i32x8 g4 = {0, 0, 0, 0, 0, 0, 0, 0};
        __builtin_amdgcn_tensor_load_to_lds(g0, g1, g2, g3, g4, 0);
#else
        __builtin_amdgcn_tensor_load_to_lds(g0, g1, g2, g3, 0);
#endif
    }

    // Small weights staged by ALU path while the TDM streams the tile.
    for (int i = tid; i < KDIM * HID; i += TPB_A)
        sB[i / HID][i % HID] = W1[i];
    if (tid < HID) { sb1[tid] = b1[tid]; sW2[tid] = W2[tid]; }

    // Warm L2 with the next block's tile for the next TDM (global_prefetch_b8).
    if ((int)blockIdx.x + 1 < (int)gridDim.x)
        __builtin_prefetch(table + (size_t)(r0 + ROWS_PER_BLOCK) * KDIM + tid * 128, 0, 1);

    if (tid < 32) __builtin_amdgcn_s_wait_tensorcnt(0);   // wave 0: DMA complete
    __syncthreads();                                      // publish LDS to all waves

    // ---- WMMA f32 16x16x4, K=128 in 32 steps, two 16-col tiles of h.
    // A layout (16x4 f32): lanes 0-15 M=0-15 hold K={0,1}; lanes 16-31 hold K={2,3}.
    const int lane = tid & 31;
    const int wid  = tid >> 5;
    const int mrow = wid * 16 + (lane & 15);
    const int ka   = (lane & 16) ? 2 : 0;
    const int ncol = lane & 15;

    v8f c0 = {};  // h columns 0..15
    v8f c1 = {};  // h columns 16..31
    #pragma unroll
    for (int kt = 0; kt < KDIM; kt += 4) {
        v2f a, bl, bh;
        a[0]  = sA[mrow][kt + ka];
        a[1]  = sA[mrow][kt + ka + 1];
        bl[0] = sB[kt + ka][ncol];
        bl[1] = sB[kt + ka + 1][ncol];
        bh[0] = sB[kt + ka][ncol + 16];
        bh[1] = sB[kt + ka + 1][ncol + 16];
        c0 = __builtin_amdgcn_wmma_f32_16x16x4_f32(false, a, false, bl, (short)0, c0, false, false);
        c1 = __builtin_amdgcn_wmma_f32_16x16x4_f32(false, a, false, bh, (short)0, c1, false, false);
    }

    // Epilogue: score[m] = sum_n relu(h[m][n]+b1[n]) * W2[n] + b2.
    // C layout: lanes 0-15 hold (M=vgpr, N=lane), lanes 16-31 hold (M=8+vgpr, N=lane-16).
    const float b1l = sb1[ncol], b1h = sb1[ncol + 16];
    const float w2l = sW2[ncol], w2h = sW2[ncol + 16];
    const float bias2 = b2[0];
    #pragma unroll
    for (int i = 0; i < 8; ++i) {
        float p = fmaxf(c0[i] + b1l, 0.0f) * w2l + fmaxf(c1[i] + b1h, 0.0f) * w2h;
        p += __shfl_xor(p, 1, 32);
        p += __shfl_xor(p, 2, 32);
        p += __shfl_xor(p, 4, 32);
        p += __shfl_xor(p, 8, 32);
        if ((lane & 15) == 0) {
            const int m = wid * 16 + ((lane & 16) ? 8 : 0) + i;
            scores[(size_t)r0 + m] = p + bias2;
        }
    }
}

// ---------------------------------------------------------------------------
// Kernel 2 (one block per batch): min/max scale, bitonic sort (descending),
// PAV isotonic-KL regression, ranks = exp(s - sol), capacity floor, scatter.
// ---------------------------------------------------------------------------
__global__ __launch_bounds__(TPB_B)
void rank_kernel(const float* __restrict__ scores,
                 const int* __restrict__ block_size,
                 float* __restrict__ out)
{
    __shared__ float sv[NROWS];
    __shared__ int   si[NROWS];
    __shared__ float ssol[NROWS];
    __shared__ float blse[NROWS];    // PAV stack: block logsumexp (also min scratch)
    __shared__ float bw[NROWS];      // PAV stack: block weight sum (also max scratch)
    __shared__ int   bstart[NROWS];  // PAV stack: block start

    const int tid = threadIdx.x;
    const int b   = blockIdx.x;
    const float* s = scores + (size_t)b * NROWS;

    float lmn = __builtin_inff(), lmx = -__builtin_inff();
    for (int i = tid; i < NROWS; i += TPB_B) {
        const float v = s[i];
        sv[i] = v; si[i] = i;
        lmn = fminf(lmn, v); lmx = fmaxf(lmx, v);
    }
    blse[tid] = lmn; bw[tid] = lmx;
    __syncthreads();
    for (int off = TPB_B / 2; off > 0; off >>= 1) {
        if (tid < off) {
            blse[tid] = fminf(blse[tid], blse[tid + off]);
            bw[tid]   = fmaxf(bw[tid],   bw[tid + off]);
        }
        __syncthreads();
    }
    const float mn = blse[0], mx = bw[0];
    const float sc = 100.0f / (mx - mn + 0.1f);
    __syncthreads();

    for (int i = tid; i < NROWS; i += TPB_B) sv[i] = (sv[i] - mn) * sc;
    __syncthreads();

    // Bitonic sort, descending, with index payload
    for (int k = 2; k <= NROWS; k <<= 1) {
        for (int j = k >> 1; j > 0; j >>= 1) {
            for (int i = tid; i < NROWS; i += TPB_B) {
                const int l = i ^ j;
                if (l > i) {
                    const bool desc = ((i & k) == 0);
                    const float a = sv[i], c = sv[l];
                    if (desc ? (a < c) : (a > c)) {
                        sv[i] = c; sv[l] = a;
                        const int t = si[i]; si[i] = si[l]; si[l] = t;
                    }
                }
            }
            __syncthreads();
        }
    }

    // PAV: exact solution of the KL isotonic problem (equals the minimax formula)
    if (tid == 0) {
        int top = -1;
        for (int i = 0; i < NROWS; ++i) {
            float lse  = sv[i];
            float wsum = (float)(NROWS - i);  // w = [n, n-1, ..., 1]
            int   st   = i;
            float val  = lse - logf(wsum);
            while (top >= 0) {
                const float pv = blse[top] - logf(bw[top]);
                if (pv >= val) break;           // non-increasing satisfied
                const float a = blse[top];      // merge with previous block
                const float m = fmaxf(a, lse);
                lse   = m + logf(expf(a - m) + expf(lse - m));
                wsum += bw[top];
                st    = bstart[top];
                val   = lse - logf(wsum);
                --top;
            }
            ++top;
            blse[top] = lse; bw[top] = wsum; bstart[top] = st;
        }
        for (int bi = 0; bi <= top; ++bi) {
            const float v = blse[bi] - logf(bw[bi]);
            const int e = (bi < top) ? bstart[bi + 1] : NROWS;
            for (int i = bstart[bi]; i < e; ++i) ssol[i] = v;
        }
    }
    __syncthreads();

    const float cap = (float)block_size[0];
    for (int i = tid; i < NROWS; i += TPB_B) {
        const float r     = expf(sv[i] - ssol[i]);
        const float other = fmodf(r, cap);
        out[(size_t)b * NROWS + si[i]] = (r - other) / cap + 1.0f;
    }
}

// ---------------------------------------------------------------------------
extern "C" void kernel_launch(void* const* d_in, const int* in_sizes, int n_in,
                              void* d_out, int out_size, void* d_ws, size_t ws_size,
                              hipStream_t stream)
{
    const float* table = (const float*)d_in[0];
    const float* W1    = (const float*)d_in[1];
    const float* b1    = (const float*)d_in[2];
    const float* W2    = (const float*)d_in[3];
    const float* b2    = (const float*)d_in[4];
    const int*   bs    = (const int*)d_in[5];
    float* out    = (float*)d_out;
    float* scores = (float*)d_ws;   // 16384 f32 = 64 KB scratch

    const int total_rows = in_sizes[0] / KDIM;          // B * rows = 16384
    mlp_scores_wmma<<<dim3(total_rows / ROWS_PER_BLOCK), dim3(TPB_A), 0, stream>>>(
        table, W1, b1, W2, b2, scores, total_rows);

    const int nbatch = total_rows / NROWS;              // 8
    rank_kernel<<<dim3(nbatch), dim3(TPB_B), 0, stream>>>(scores, bs, out);
}